// RNNModel_34866544509095
// MI455X (gfx1250) — compile-verified
//
#include <hip/hip_runtime.h>
#include <math.h>

typedef float v2f __attribute__((ext_vector_type(2)));
typedef float v8f __attribute__((ext_vector_type(8)));

#define H_SZ 50
#define O_OUT 10
#define B_SZ 512
#define T_STEPS 2048
#define RS 68      // padded row stride in floats (8B aligned, bank-conflict free)
#define XBLK 64    // timesteps of x staged per chunk
#define NT 4       // N tiles (64/16)
#define KC 16      // K chunks (64/4)

// Native CDNA5 tanh (V_TANH_F32, single TRANS32 op) when the builtin exists;
// otherwise a branch-free native-exp formulation (saturates to +-1 correctly).
__device__ __forceinline__ float fast_tanh(float v) {
#if __has_builtin(__builtin_amdgcn_tanhf)
    return __builtin_amdgcn_tanhf(v);
#else
    float e = __expf(v + v);                    // v_exp_f32 (native)
    return 1.0f - __fdividef(2.0f, e + 1.0f);   // rcp-based, no branches
#endif
}

#define WMMA_F32(A, B, C) \
    __builtin_amdgcn_wmma_f32_16x16x4_f32(false, (A), false, (B), (short)0, (C), false, false)

__global__ __launch_bounds__(128, 1) void rnn2_tanh_wmma(
    const float* __restrict__ x,
    const float* __restrict__ W_ih0, const float* __restrict__ W_hh0,
    const float* __restrict__ b_ih0, const float* __restrict__ b_hh0,
    const float* __restrict__ W_ih1, const float* __restrict__ W_hh1,
    const float* __restrict__ b_ih1, const float* __restrict__ b_hh1,
    const float* __restrict__ W_fc,  const float* __restrict__ b_fc,
    float* __restrict__ out)
{
    // B-fragment-swizzled weights (built once, then hoisted to registers)
    __shared__ float wB0[NT][KC][32][2];   // W_hh0
    __shared__ float wB1[NT][KC][32][2];   // W_ih1
    __shared__ float wB2[NT][KC][32][2];   // W_hh1
    __shared__ float h0buf[2][16 * RS];    // hidden state double buffers, [m][k]
    __shared__ float h1buf[2][16 * RS];
    __shared__ float xs[16 * RS];          // staged x block: [m][t mod 64]

    const int tid  = threadIdx.x;
    const int lane = tid & 31;
    const int wv   = tid >> 5;     // wave id = N-tile id
    const int r    = lane & 15;
    const int hf   = lane >> 4;
    const int b0   = blockIdx.x * 16;

    // ---- init: zero hidden buffers, build weight fragments ----
    {
        float* h0f = &h0buf[0][0];
        float* h1f = &h1buf[0][0];
        for (int i = tid; i < 2 * 16 * RS; i += 128) { h0f[i] = 0.f; h1f[i] = 0.f; }
        float* w0f = &wB0[0][0][0][0];
        float* w1f = &wB1[0][0][0][0];
        float* w2f = &wB2[0][0][0][0];
        for (int idx = tid; idx < NT * KC * 32 * 2; idx += 128) {
            int v  = idx & 1;
            int ln = (idx >> 1) & 31;
            int kc = (idx >> 6) & 15;
            int nt = idx >> 10;
            int n  = nt * 16 + (ln & 15);
            int k  = kc * 4 + 2 * (ln >> 4) + v;
            float a = 0.f, b = 0.f, c = 0.f;
            if (n < H_SZ && k < H_SZ) {
                a = W_hh0[n * H_SZ + k];   // B[k][n] = W[n][k]  (h @ W^T)
                b = W_ih1[n * H_SZ + k];
                c = W_hh1[n * H_SZ + k];
            }
            w0f[idx] = a; w1f[idx] = b; w2f[idx] = c;
        }
    }

    // per-lane constants for this lane's fixed output column n
    const int n = wv * 16 + r;
    float wih = 0.f, bias0n = 0.f, bias1n = 0.f;
    if (n < H_SZ) {
        wih    = W_ih0[n];                    // (H,1)
        bias0n = b_ih0[n] + b_hh0[n];
        bias1n = b_ih1[n] + b_hh1[n];
    }
    __syncthreads();

    // ---- hoist this wave's B fragments into registers (loop-invariant) ----
    v2f rB0[KC], rB1[KC], rB2[KC];
    #pragma unroll
    for (int kc = 0; kc < KC; ++kc) {
        rB0[kc] = *(const v2f*)&wB0[wv][kc][lane][0];
        rB1[kc] = *(const v2f*)&wB1[wv][kc][lane][0];
        rB2[kc] = *(const v2f*)&wB2[wv][kc][lane][0];
    }

    int p = 0;
    for (int t = 0; t < T_STEPS; ++t) {
        const int tt = t & (XBLK - 1);
        if (tt == 0) {
            // stage x[b0..b0+15][t..t+63]  (coalesced along t)
            for (int i = tid; i < 16 * XBLK; i += 128) {
                int rr = i >> 6;
                int cc = i & (XBLK - 1);
                xs[rr * RS + cc] = x[(size_t)(b0 + rr) * T_STEPS + t + cc];
            }
            __syncthreads();
        }

        // ---------- layer 0: d0 = tanh(x*Wih0 + b0 + h0_prev @ Whh0^T) ----------
        // two independent WMMA chains (even/odd K-chunks) for pipelining
        v8f acc0, acc1;
        #pragma unroll
        for (int v = 0; v < 8; ++v) {
            float xm = xs[(v + 8 * hf) * RS + tt];      // broadcast read
            acc0[v] = fmaf(xm, wih, bias0n);
            acc1[v] = 0.0f;
        }
        const float* h0r = &h0buf[p][0];
        #pragma unroll
        for (int kc = 0; kc < KC; kc += 2) {
            v2f aA = *(const v2f*)&h0r[r * RS + kc * 4 + 2 * hf];
            v2f aB = *(const v2f*)&h0r[r * RS + (kc + 1) * 4 + 2 * hf];
            acc0 = WMMA_F32(aA, rB0[kc],     acc0);
            acc1 = WMMA_F32(aB, rB0[kc + 1], acc1);
        }
        v8f c0 = acc0 + acc1;
        float* h0w = &h0buf[1 - p][0];
        #pragma unroll
        for (int v = 0; v < 8; ++v)
            h0w[(v + 8 * hf) * RS + wv * 16 + r] = fast_tanh(c0[v]);
        __syncthreads();

        // ---------- layer 1: d1 = tanh(h0_new @ Wih1^T + h1_prev @ Whh1^T + b1) ----------
        // one independent chain per GEMM
        v8f acc2, acc3;
        #pragma unroll
        for (int v = 0; v < 8; ++v) { acc2[v] = bias1n; acc3[v] = 0.0f; }
        const float* h1r = &h1buf[p][0];
        #pragma unroll
        for (int kc = 0; kc < KC; ++kc) {
            v2f a0 = *(const v2f*)&h0w[r * RS + kc * 4 + 2 * hf];
            acc2 = WMMA_F32(a0, rB1[kc], acc2);
            v2f a1 = *(const v2f*)&h1r[r * RS + kc * 4 + 2 * hf];
            acc3 = WMMA_F32(a1, rB2[kc], acc3);
        }
        v8f c1 = acc2 + acc3;
        float* h1w = &h1buf[1 - p][0];
        #pragma unroll
        for (int v = 0; v < 8; ++v)
            h1w[(v + 8 * hf) * RS + wv * 16 + r] = fast_tanh(c1[v]);
        __syncthreads();

        p ^= 1;
    }

    // ---------- FC: out = h1_T @ W_fc^T + b_fc  (tiny, scalar) ----------
    const float* h1f = &h1buf[p][0];
    for (int i = tid; i < 16 * O_OUT; i += 128) {
        int m = i / O_OUT, o = i % O_OUT;
        float acc = b_fc[o];
        #pragma unroll 5
        for (int k = 0; k < H_SZ; ++k)
            acc = fmaf(h1f[m * RS + k], W_fc[o * H_SZ + k], acc);
        out[(size_t)(b0 + m) * O_OUT + o] = acc;
    }
}

extern "C" void kernel_launch(void* const* d_in, const int* in_sizes, int n_in,
                              void* d_out, int out_size, void* d_ws, size_t ws_size,
                              hipStream_t stream) {
    const float* x     = (const float*)d_in[0];
    const float* W_ih0 = (const float*)d_in[1];
    const float* W_hh0 = (const float*)d_in[2];
    const float* b_ih0 = (const float*)d_in[3];
    const float* b_hh0 = (const float*)d_in[4];
    const float* W_ih1 = (const float*)d_in[5];
    const float* W_hh1 = (const float*)d_in[6];
    const float* b_ih1 = (const float*)d_in[7];
    const float* b_hh1 = (const float*)d_in[8];
    const float* W_fc  = (const float*)d_in[9];
    const float* b_fc  = (const float*)d_in[10];
    float* out = (float*)d_out;

    rnn2_tanh_wmma<<<B_SZ / 16, 128, 0, stream>>>(
        x, W_ih0, W_hh0, b_ih0, b_hh0, W_ih1, W_hh1, b_ih1, b_hh1, W_fc, b_fc, out);
}